// RippleAdderTile_36369783063301
// MI455X (gfx1250) — compile-verified
//
#include <hip/hip_runtime.h>
#include <stdint.h>

// out = a + b (the bit-serial ripple adder reduces exactly to integer add,
// exact in f32 since a,b < 2^24).  Computed on the matrix pipe as
// D = sum_j A_j x B_j + C with A = I16 (split into 4 K-slices), B = b-tile,
// C = a-tile, via 4 chained V_WMMA_F32_16X16X4_F32 per 16x16 tile.
//
// Operand layouts (wave32):
//   A: 16x4 f32  -> 2 VGPRs: lane half hi: VGPR r holds K = 2*hi + r, M = lane%16
//   B:  4x16 f32 -> 2 VGPRs: VGPR r, half hi: row K = 2*hi + r, col N = lane%16
//   C/D: 16x16   -> 8 VGPRs: VGPR r: (M = r + 8*hi, N = lane%16)
// Flat map f(M,N) = (M/8)*128 + N*8 + M%8  =>  lane l owns floats [8l, 8l+8)
// of the C/D tile: a-loads and out-stores are contiguous B128.
typedef float v2f __attribute__((ext_vector_type(2)));
typedef float v8f __attribute__((ext_vector_type(8)));

#define TILE 256            // 16x16 f32 per wave32
#define WAVES_PER_BLOCK 8
#define TILES_PER_WAVE 4

__global__ __launch_bounds__(256) void ripple_add_wmma(const float* __restrict__ a,
                                                       const float* __restrict__ b,
                                                       float* __restrict__ out,
                                                       int n) {
    // double-buffered b staging: 8 waves x 2 bufs x 1KB
    __shared__ float ldsB[WAVES_PER_BLOCK * 2 * TILE];

    const int lane = threadIdx.x & 31;
    const int wave = (int)(threadIdx.x >> 5);
    const long firstTile =
        ((long)blockIdx.x * WAVES_PER_BLOCK + wave) * TILES_PER_WAVE;
    const long start = firstTile * TILE;

    if (start + (long)TILES_PER_WAVE * TILE <= (long)n) {
        float* wLds = &ldsB[wave * 2 * TILE];
        const uint32_t ldsBase = (uint32_t)(uintptr_t)wLds + (uint32_t)(lane * 16);

        // ---- identity A_j (columns 4j..4j+3 of I16), built once ----
        const int hi = lane >> 4;
        const int m  = lane & 15;
        v2f A[4];
#pragma unroll
        for (int j = 0; j < 4; ++j)
#pragma unroll
            for (int r = 0; r < 2; ++r)
                A[j][r] = (m == 4 * j + 2 * hi + r) ? 1.0f : 0.0f;

        // ---- prologue: async-stage b tile 0 into buffer 0 ----
        {
            uint64_t gAddr = (uint64_t)(uintptr_t)(b + start) + (uint64_t)(lane * 16);
            asm volatile(
                "global_load_async_to_lds_b128 %0, %1, off\n\t"
                "global_load_async_to_lds_b128 %0, %1, off offset:512"
                :
                : "v"(ldsBase), "v"(gAddr)
                : "memory");
        }

#pragma unroll
        for (int t = 0; t < TILES_PER_WAVE; ++t) {
            const long tbase = start + (long)t * TILE;

            // ---- prefetch next tile's b into the other buffer ----
            if (t + 1 < TILES_PER_WAVE) {
                uint32_t ldsNext = ldsBase + (uint32_t)(((t + 1) & 1) * TILE * 4);
                uint64_t gNext   = (uint64_t)(uintptr_t)(b + tbase + TILE) +
                                   (uint64_t)(lane * 16);
                asm volatile(
                    "global_load_async_to_lds_b128 %0, %1, off\n\t"
                    "global_load_async_to_lds_b128 %0, %1, off offset:512"
                    :
                    : "v"(ldsNext), "v"(gNext)
                    : "memory");
            }

            // ---- C accumulator = a tile (two contiguous B128 per lane) ----
            v8f c = *reinterpret_cast<const v8f*>(a + tbase + (long)lane * 8);

            // ---- wait for current tile's copy; keep the prefetch in flight ----
            if (t + 1 < TILES_PER_WAVE)
                asm volatile("s_wait_asynccnt 0x2" ::: "memory");
            else
                asm volatile("s_wait_asynccnt 0x0" ::: "memory");

            // ---- gather B_j rows from LDS (scatter absorbed by LDS banks) ----
            const float* buf = wLds + (t & 1) * TILE;
            v2f B[4];
#pragma unroll
            for (int j = 0; j < 4; ++j)
#pragma unroll
                for (int r = 0; r < 2; ++r) {
                    const int M = 4 * j + 2 * hi + r;                   // b-tile row
                    const int f = ((M >> 3) << 7) + (m << 3) + (M & 7); // flat map
                    B[j][r] = buf[f];
                }

            // ---- 4 chained WMMAs: c += A_j x B_j  (== a+b, exact) ----
#pragma unroll
            for (int j = 0; j < 4; ++j) {
                c = __builtin_amdgcn_wmma_f32_16x16x4_f32(
                        /*neg_a=*/false, A[j],
                        /*neg_b=*/false, B[j],
                        /*c_mod=*/(short)0, c,
                        /*reuse_a=*/false, /*reuse_b=*/false);
            }

            // ---- store D: two contiguous B128 per lane ----
            *reinterpret_cast<v8f*>(out + tbase + (long)lane * 8) = c;
        }
    } else {
        // Tail (unused for N = 4M, tile-divisible): scalar exact add.
        const long stop = start + (long)TILES_PER_WAVE * TILE;
        for (long i = start + lane; i < (long)n && i < stop; i += 32)
            out[i] = a[i] + b[i];
    }
}

extern "C" void kernel_launch(void* const* d_in, const int* in_sizes, int n_in,
                              void* d_out, int out_size, void* d_ws, size_t ws_size,
                              hipStream_t stream) {
    const float* a = (const float*)d_in[0];
    const float* b = (const float*)d_in[1];
    float* out = (float*)d_out;
    const int n = in_sizes[0];

    const int elems_per_block = TILE * WAVES_PER_BLOCK * TILES_PER_WAVE; // 8192
    const int blocks = (n + elems_per_block - 1) / elems_per_block;
    ripple_add_wmma<<<blocks, 256, 0, stream>>>(a, b, out, n);
}